// KGModel_37263136260513
// MI455X (gfx1250) — compile-verified
//
#include <hip/hip_runtime.h>

typedef __attribute__((ext_vector_type(2))) float v2f;
typedef __attribute__((ext_vector_type(8))) float v8f;

#define HD 128  // hidden dim

// ---------------- degree / normalization precompute ----------------

__global__ __launch_bounds__(256) void init_deg_kernel(float* deg, int n) {
  int i = blockIdx.x * 256 + threadIdx.x;
  if (i < n) deg[i] = 1.0f;  // self loop contributes 1 to in-degree
}

__global__ __launch_bounds__(256) void deg_edge_kernel(const int* __restrict__ dst,
                                                       int e, float* deg) {
  int i = blockIdx.x * 256 + threadIdx.x;
  if (i < e) atomicAdd(&deg[dst[i]], 1.0f);
}

__global__ __launch_bounds__(256) void dinv_kernel(float* deg, int n) {
  int i = blockIdx.x * 256 + threadIdx.x;
  if (i < n) deg[i] = rsqrtf(deg[i]);  // deg >= 1 always (self loops)
}

__global__ __launch_bounds__(256) void zero_kernel(float4* p, long n4) {
  long i = (long)blockIdx.x * 256 + threadIdx.x;
  if (i < n4) p[i] = make_float4(0.f, 0.f, 0.f, 0.f);
}

// ---------------- fp32 WMMA GEMM: Out = f(X @ W [+bias][+res1][+res2]) ----------------
// X: [nrows,128], W: [128,128] row-major (h[n] = sum_k x[k]*W[k*128+n]).
// One wave32 computes a full 16x128 output strip: 8 accumulator tiles, so each
// A fragment (scattered global load) is loaded once per K-step and reused by
// 8 V_WMMA_F32_16X16X4_F32 ops whose B fragments come from LDS-resident W.
// K=128 -> 32 K-steps -> 256 wmma ops per wave.

__global__ __launch_bounds__(128) void gemm128_wmma_kernel(
    const float* __restrict__ X, const float* __restrict__ W,
    const float* __restrict__ bias,
    const float* __restrict__ res1, const float* __restrict__ res2,
    float* __restrict__ Out, int nrows, int do_relu)
{
  __shared__ float sW[HD * HD];  // 64 KB of the 320 KB WGP LDS
  for (int i = threadIdx.x * 4; i < HD * HD; i += 128 * 4)
    *(float4*)(sW + i) = *(const float4*)(W + i);
  __syncthreads();

  int lane = threadIdx.x & 31;
  int wave = threadIdx.x >> 5;
  long mTile = (long)blockIdx.x * 4 + wave;   // one 16-row strip per wave
  long mTiles = (nrows + 15) >> 4;
  if (mTile >= mTiles) return;                // whole-wave uniform predicate

  int l16  = lane & 15;
  int koff = (lane >> 4) << 1;        // lanes 0-15: K=0,1 ; lanes 16-31: K=2,3
  long row = mTile * 16 + l16;        // A row (M)
  const float* xrow = X + row * HD;

  v8f c[8];
#pragma unroll
  for (int t = 0; t < 8; ++t) c[t] = (v8f){};

#pragma unroll
  for (int k = 0; k < HD; k += 4) {
    v2f a;
    float2 av = *(const float2*)(xrow + k + koff);   // 8B-aligned (k+koff even)
    a.x = av.x; a.y = av.y;
#pragma unroll
    for (int t = 0; t < 8; ++t) {
      int col = t * 16 + l16;                         // B/C column (N)
      v2f b;
      b.x = sW[(k + koff) * HD + col];
      b.y = sW[(k + koff + 1) * HD + col];
      c[t] = __builtin_amdgcn_wmma_f32_16x16x4_f32(
          /*neg_a=*/false, a, /*neg_b=*/false, b,
          /*c_mod=*/(short)0, c[t], /*reuse_a=*/false, /*reuse_b=*/false);
    }
  }

  // C layout: VGPR r holds M = r + (lane>=16 ? 8 : 0), N = lane%16
  int rbase = (lane >> 4) << 3;  // 0 or 8
#pragma unroll
  for (int t = 0; t < 8; ++t) {
    int col = t * 16 + l16;
    float bc = bias ? bias[col] : 0.0f;
#pragma unroll
    for (int r = 0; r < 8; ++r) {
      long m = mTile * 16 + rbase + r;
      long idx = m * HD + col;
      float v = c[t][r] + bc;
      if (res1) v += res1[idx];
      if (res2) v += res2[idx];
      if (do_relu) v = fmaxf(v, 0.0f);
      Out[idx] = v;
    }
  }
}

// ---------------- edge scatter: acc[dst] += dinv[src]*dinv[dst] * h[src] ----------------
// One wave32 per edge; lane handles 4 consecutive channels (coalesced 512B row).

__global__ __launch_bounds__(256) void scatter_kernel(
    const int* __restrict__ src, const int* __restrict__ dst, int e,
    const float* __restrict__ dinv, const float* __restrict__ h,
    float* __restrict__ acc)
{
  long gw = ((long)blockIdx.x * 256 + threadIdx.x) >> 5;
  int lane = threadIdx.x & 31;
  if (gw >= e) return;
  int s = src[gw];
  int d = dst[gw];
  float nrm = dinv[s] * dinv[d];
  const float4 hv = *(const float4*)(h + (long)s * HD + lane * 4);
  float* out = acc + (long)d * HD + lane * 4;
  atomicAdd(out + 0, hv.x * nrm);
  atomicAdd(out + 1, hv.y * nrm);
  atomicAdd(out + 2, hv.z * nrm);
  atomicAdd(out + 3, hv.w * nrm);
}

// ---------------- fused self-loop + bias + layernorm (+relu) ----------------
// out[v] = f( LN( acc[v] + dinv[v]^2 * h[v] + b ) * g + be ); one wave per row.

__global__ __launch_bounds__(256) void ln_kernel(
    const float* __restrict__ acc, const float* __restrict__ h,
    const float* __restrict__ dinv, const float* __restrict__ bias,
    const float* __restrict__ g, const float* __restrict__ be,
    float* __restrict__ out, int nrows, int do_relu)
{
  long row = ((long)blockIdx.x * 256 + threadIdx.x) >> 5;
  int lane = threadIdx.x & 31;
  if (row >= nrows) return;
  float di = dinv[row];
  float sl = di * di;  // self-loop norm
  long base = row * HD + lane * 4;
  float4 a  = *(const float4*)(acc + base);
  float4 hv = *(const float4*)(h + base);
  float4 bb = *(const float4*)(bias + lane * 4);
  float v0 = a.x + sl * hv.x + bb.x;
  float v1 = a.y + sl * hv.y + bb.y;
  float v2 = a.z + sl * hv.z + bb.z;
  float v3 = a.w + sl * hv.w + bb.w;

  float s = v0 + v1 + v2 + v3;
  for (int m = 16; m >= 1; m >>= 1) s += __shfl_xor(s, m);
  float mu = s * (1.0f / 128.0f);
  float d0 = v0 - mu, d1 = v1 - mu, d2 = v2 - mu, d3 = v3 - mu;
  float ss = d0 * d0 + d1 * d1 + d2 * d2 + d3 * d3;
  for (int m = 16; m >= 1; m >>= 1) ss += __shfl_xor(ss, m);
  float inv = rsqrtf(ss * (1.0f / 128.0f) + 1e-5f);

  float4 gg = *(const float4*)(g + lane * 4);
  float4 ee = *(const float4*)(be + lane * 4);
  float o0 = d0 * inv * gg.x + ee.x;
  float o1 = d1 * inv * gg.y + ee.y;
  float o2 = d2 * inv * gg.z + ee.z;
  float o3 = d3 * inv * gg.w + ee.w;
  if (do_relu) {
    o0 = fmaxf(o0, 0.f); o1 = fmaxf(o1, 0.f);
    o2 = fmaxf(o2, 0.f); o3 = fmaxf(o3, 0.f);
  }
  *(float4*)(out + base) = make_float4(o0, o1, o2, o3);
}

// ---------------- driver ----------------

extern "C" void kernel_launch(void* const* d_in, const int* in_sizes, int n_in,
                              void* d_out, int out_size, void* d_ws, size_t ws_size,
                              hipStream_t stream) {
  const int* edge = (const int*)d_in[0];
  int E = in_sizes[0] / 2;
  const int* src = edge;
  const int* dst = edge + E;
  const float* x0 = (const float*)d_in[1];
  int N = in_sizes[1] / HD;
  const float* W1  = (const float*)d_in[2];
  const float* b1  = (const float*)d_in[3];
  const float* g1  = (const float*)d_in[4];
  const float* be1 = (const float*)d_in[5];
  const float* W2  = (const float*)d_in[6];
  const float* b2  = (const float*)d_in[7];
  const float* g2  = (const float*)d_in[8];
  const float* be2 = (const float*)d_in[9];
  const float* W3  = (const float*)d_in[10];
  const float* b3  = (const float*)d_in[11];
  const float* g3  = (const float*)d_in[12];
  const float* be3 = (const float*)d_in[13];
  const float* Wt1 = (const float*)d_in[14];
  const float* bt1 = (const float*)d_in[15];
  const float* Wt2 = (const float*)d_in[16];
  const float* bt2 = (const float*)d_in[17];
  float* out = (float*)d_out;

  // workspace layout (256B aligned slabs)
  char* ws = (char*)d_ws;
  size_t nd = (size_t)N * HD;
  size_t off = 0;
  float* dinv = (float*)(ws + off); off += ((size_t)N * 4 + 255) & ~(size_t)255;
  float* h    = (float*)(ws + off); off += nd * 4;
  float* acc  = (float*)(ws + off); off += nd * 4;
  float* xa   = (float*)(ws + off); off += nd * 4;
  float* x3   = (float*)(ws + off); off += nd * 4;
  (void)ws_size; (void)n_in; (void)out_size;

  int nb = (N + 255) / 256;
  long n4 = (long)(nd / 4);
  int zb  = (int)((n4 + 255) / 256);
  int sb  = (E + 7) / 8;                 // 8 edge-waves per 256-thread block
  int lnb = (N + 7) / 8;                 // 8 row-waves per 256-thread block
  long mTiles = (N + 15) / 16;
  int gb = (int)((mTiles + 3) / 4);      // 4 strip-waves per 128-thread block

  // symmetric normalization dinv = rsqrt(deg)
  init_deg_kernel<<<nb, 256, 0, stream>>>(dinv, N);
  deg_edge_kernel<<<(E + 255) / 256, 256, 0, stream>>>(dst, E, dinv);
  dinv_kernel<<<nb, 256, 0, stream>>>(dinv, N);

  // layer 1: h = x0@W1 ; acc = scatter ; xa = relu(LN(acc + dinv^2*h + b1))
  gemm128_wmma_kernel<<<gb, 128, 0, stream>>>(x0, W1, nullptr, nullptr, nullptr, h, N, 0);
  zero_kernel<<<zb, 256, 0, stream>>>((float4*)acc, n4);
  scatter_kernel<<<sb, 256, 0, stream>>>(src, dst, E, dinv, h, acc);
  ln_kernel<<<lnb, 256, 0, stream>>>(acc, h, dinv, b1, g1, be1, xa, N, 1);

  // layer 2
  gemm128_wmma_kernel<<<gb, 128, 0, stream>>>(xa, W2, nullptr, nullptr, nullptr, h, N, 0);
  zero_kernel<<<zb, 256, 0, stream>>>((float4*)acc, n4);
  scatter_kernel<<<sb, 256, 0, stream>>>(src, dst, E, dinv, h, acc);
  ln_kernel<<<lnb, 256, 0, stream>>>(acc, h, dinv, b2, g2, be2, xa, N, 1);

  // layer 3 (no relu)
  gemm128_wmma_kernel<<<gb, 128, 0, stream>>>(xa, W3, nullptr, nullptr, nullptr, h, N, 0);
  zero_kernel<<<zb, 256, 0, stream>>>((float4*)acc, n4);
  scatter_kernel<<<sb, 256, 0, stream>>>(src, dst, E, dinv, h, acc);
  ln_kernel<<<lnb, 256, 0, stream>>>(acc, h, dinv, b3, g3, be3, x3, N, 0);

  // transform MLP + residual: out = relu(x3@Wt1+bt1)@Wt2 + bt2 + x3 + x0
  gemm128_wmma_kernel<<<gb, 128, 0, stream>>>(x3, Wt1, bt1, nullptr, nullptr, h, N, 1);
  gemm128_wmma_kernel<<<gb, 128, 0, stream>>>(h, Wt2, bt2, x3, x0, out, N, 0);
}